// GraphSage_26164940767482
// MI455X (gfx1250) — compile-verified
//
#include <hip/hip_runtime.h>

// GraphSAGE (3-layer SAGEConv, mean aggregation) for MI455X / gfx1250.
//   - edge aggregation: float4 gathers + native global_atomic_add_f32
//     (whole activation set is resident in the 192MB L2)
//   - dense layers: V_WMMA_F32_16X16X4_F32 (full fp32, matches reference),
//     with weights pre-packed into the WMMA B-fragment layout so the inner
//     loop is 2x global_load_b64 + v_pk_mul_f32 + v_wmma per K-step
//     (no lane-dependent branches, no 64-bit muls).

typedef __attribute__((ext_vector_type(2))) float v2f;
typedef __attribute__((ext_vector_type(8))) float v8f;

// ---------------------------------------------------------------- utilities

__global__ void zero_f32_kernel(float* __restrict__ p, long long n) {
    long long i = (long long)blockIdx.x * blockDim.x + threadIdx.x;
    long long stride = (long long)gridDim.x * blockDim.x;
    for (; i < n; i += stride) p[i] = 0.0f;
}

__global__ void degree_kernel(const int* __restrict__ dst, int* __restrict__ cnt, int E) {
    int e = blockIdx.x * blockDim.x + threadIdx.x;
    if (e < E) atomicAdd(&cnt[dst[e]], 1);
}

__global__ void invdeg_kernel(const int* __restrict__ cnt, float* __restrict__ inv, int N) {
    int i = blockIdx.x * blockDim.x + threadIdx.x;
    if (i < N) {
        int c = cnt[i];
        inv[i] = 1.0f / (float)(c > 0 ? c : 1);
    }
}

// -------------------------------------------------- weight / bias packing
// Pack W[K,Nout] row-major into B-fragment layout, zero-padded to Np cols:
//   Bp (as float2)[kp*Np + col] = { W[2kp][col], W[2kp+1][col] },  kp = 0..K/2-1
// In the GEMM, lane (kh,col) reads float2 at (kh*Np + col) and strides 2*Np
// per K4 step -> a single coalesced global_load_b64, no masking needed.

__global__ void pack_weights_kernel(const float* __restrict__ W,
                                    float* __restrict__ Bp,  // [K/2 * Np * 2] floats
                                    int K, int Nout, int Np) {
    int tid = blockIdx.x * blockDim.x + threadIdx.x;
    int total = (K / 2) * Np;
    if (tid >= total) return;
    int kp  = tid / Np;
    int col = tid - kp * Np;
    float x0 = (col < Nout) ? W[(2 * kp) * Nout + col]     : 0.0f;
    float x1 = (col < Nout) ? W[(2 * kp + 1) * Nout + col] : 0.0f;
    float* d = Bp + (size_t)tid * 2;
    d[0] = x0;
    d[1] = x1;
}

__global__ void pack_bias_kernel(const float* __restrict__ b, float* __restrict__ bp,
                                 int Nout, int Np) {
    int i = blockIdx.x * blockDim.x + threadIdx.x;
    if (i < Np) bp[i] = (i < Nout) ? b[i] : 0.0f;
}

// ------------------------------------------------------------ edge scatter

__global__ void scatter_add_kernel(const float* __restrict__ h,
                                   const int* __restrict__ src,
                                   const int* __restrict__ dst,
                                   float* __restrict__ agg,
                                   long long total, int D, int lgc) {
    long long tid = (long long)blockIdx.x * blockDim.x + threadIdx.x;
    if (tid >= total) return;
    int e = (int)(tid >> lgc);
    int c = (int)(tid & ((1 << lgc) - 1));
    int s = src[e];
    int d = dst[e];
    float4 v = *((const float4*)(h + (size_t)s * D) + c);
    float* p = agg + (size_t)d * D + 4 * c;
    unsafeAtomicAdd(p + 0, v.x);
    unsafeAtomicAdd(p + 1, v.y);
    unsafeAtomicAdd(p + 2, v.z);
    unsafeAtomicAdd(p + 3, v.w);
}

// -------------------------------------------------------- fused SAGE layer
// out = act( (agg * inv_deg) @ Wl + bias + h @ Wr ), one wave32 per 16x16 tile.

__global__ void sage_wmma_kernel(const float* __restrict__ agg,     // [N,K]
                                 const float* __restrict__ inv_deg, // [N]
                                 const float* __restrict__ h,       // [N,K]
                                 const float* __restrict__ Bpl,     // packed Wl
                                 const float* __restrict__ Bpr,     // packed Wr
                                 const float* __restrict__ biasp,   // [Np] padded
                                 float* __restrict__ out,           // [N,Nout]
                                 int N, int K, int Nout, int ntn, int relu) {
    int wave = blockIdx.x * (blockDim.x >> 5) + (threadIdx.x >> 5);
    int lane = threadIdx.x & 31;
    int total_tiles = (N >> 4) * ntn;
    if (wave >= total_tiles) return;          // uniform per wave: EXEC stays all-1s

    int tm = wave / ntn;
    int tn = wave - tm * ntn;
    int Np = ntn * 16;

    int row = tm * 16 + (lane & 15);          // A-matrix row for this lane
    int col = tn * 16 + (lane & 15);          // B/C/D column for this lane
    int kh  = lane >> 4;                      // K-half: lanes 16-31 hold K+2,K+3

    v8f c = {0.f, 0.f, 0.f, 0.f, 0.f, 0.f, 0.f, 0.f};

    float idg = inv_deg[row];
    int K4 = K >> 2;

    // mean(neigh) @ Wl  (mean-division fused into the A fragment)
    {
        const v2f* pa = (const v2f*)(agg + (size_t)row * K) + kh;
        const v2f* pb = (const v2f*)Bpl + (size_t)kh * Np + col;
#pragma unroll 8
        for (int s = 0; s < K4; ++s) {
            v2f a = *pa;
            a = a * idg;                      // v_pk_mul_f32, co-executes with WMMA
            v2f b = *pb;
            c = __builtin_amdgcn_wmma_f32_16x16x4_f32(false, a, false, b,
                                                      (short)0, c, false, false);
            pa += 2;
            pb += 2 * Np;
        }
    }
    // h @ Wr
    {
        const v2f* pa = (const v2f*)(h + (size_t)row * K) + kh;
        const v2f* pb = (const v2f*)Bpr + (size_t)kh * Np + col;
#pragma unroll 8
        for (int s = 0; s < K4; ++s) {
            v2f a = *pa;
            v2f b = *pb;
            c = __builtin_amdgcn_wmma_f32_16x16x4_f32(false, a, false, b,
                                                      (short)0, c, false, false);
            pa += 2;
            pb += 2 * Np;
        }
    }

    // Epilogue: C/D layout -> VGPR v holds row (v + 8*kh), col = lane&15.
    float bv = biasp[col];                    // padded: unconditional
    int rbase = tm * 16 + 8 * kh;
    bool colok = (col < Nout);                // diverges only for the 6-wide layer
#pragma unroll
    for (int v = 0; v < 8; ++v) {
        float val = c[v] + bv;
        if (relu) val = fmaxf(val, 0.0f);
        if (colok) out[(size_t)(rbase + v) * Nout + col] = val;
    }
}

// ------------------------------------------------------------------ driver

extern "C" void kernel_launch(void* const* d_in, const int* in_sizes, int n_in,
                              void* d_out, int out_size, void* d_ws, size_t ws_size,
                              hipStream_t stream) {
    const int D_IN = 128, D_HID = 64, D_OUT = 6;
    const int N = in_sizes[0] / D_IN;      // 50000
    const int E = in_sizes[1] / 2;         // 800000

    const float* x    = (const float*)d_in[0];
    const int*   ei   = (const int*)d_in[1];
    const float* W1l  = (const float*)d_in[2];
    const float* b1   = (const float*)d_in[3];
    const float* W1r  = (const float*)d_in[4];
    const float* W2l  = (const float*)d_in[5];
    const float* b2   = (const float*)d_in[6];
    const float* W2r  = (const float*)d_in[7];
    const float* W3l  = (const float*)d_in[8];
    const float* b3   = (const float*)d_in[9];
    const float* W3r  = (const float*)d_in[10];
    float* out = (float*)d_out;

    const int* src = ei;
    const int* dst = ei + E;

    // Workspace layout (floats, all chunks even-sized -> 8B alignment holds):
    float* ws      = (float*)d_ws;
    float* inv_deg = ws;                                  // N
    float* agg     = inv_deg + N;                         // N*128
    float* h1      = agg + (size_t)N * D_IN;              // N*64
    float* h2      = h1 + (size_t)N * D_HID;              // N*64
    int*   cnt     = (int*)(h2 + (size_t)N * D_HID);      // N ints
    float* wpack   = (float*)(cnt + N);
    // packed weights: L1 (K=128,Np=64): 2*8192; L2 (64,64): 2*4096; L3 (64,16): 2*1024
    float* P1l = wpack;              float* P1r = P1l + 128 * 64;
    float* P2l = P1r + 128 * 64;     float* P2r = P2l + 64 * 64;
    float* P3l = P2r + 64 * 64;      float* P3r = P3l + 64 * 16;
    float* bp1 = P3r + 64 * 16;      // 64
    float* bp2 = bp1 + 64;           // 64
    float* bp3 = bp2 + 64;           // 16

    const int ZB = 2048, ZT = 256;

    // --- weight/bias packing (tiny) ---
    pack_weights_kernel<<<(128 / 2 * 64 + 255) / 256, 256, 0, stream>>>(W1l, P1l, 128, 64, 64);
    pack_weights_kernel<<<(128 / 2 * 64 + 255) / 256, 256, 0, stream>>>(W1r, P1r, 128, 64, 64);
    pack_weights_kernel<<<(64 / 2 * 64 + 255) / 256, 256, 0, stream>>>(W2l, P2l, 64, 64, 64);
    pack_weights_kernel<<<(64 / 2 * 64 + 255) / 256, 256, 0, stream>>>(W2r, P2r, 64, 64, 64);
    pack_weights_kernel<<<(64 / 2 * 16 + 255) / 256, 256, 0, stream>>>(W3l, P3l, 64, 6, 16);
    pack_weights_kernel<<<(64 / 2 * 16 + 255) / 256, 256, 0, stream>>>(W3r, P3r, 64, 6, 16);
    pack_bias_kernel<<<1, 64, 0, stream>>>(b1, bp1, 64, 64);
    pack_bias_kernel<<<1, 64, 0, stream>>>(b2, bp2, 64, 64);
    pack_bias_kernel<<<1, 64, 0, stream>>>(b3, bp3, 6, 16);

    // --- in-degree -> inv_deg ---
    zero_f32_kernel<<<ZB, ZT, 0, stream>>>((float*)cnt, (long long)N);
    degree_kernel<<<(E + 255) / 256, 256, 0, stream>>>(dst, cnt, E);
    invdeg_kernel<<<(N + 255) / 256, 256, 0, stream>>>(cnt, inv_deg, N);

    // --- layer 1: x (K=128) -> h1 (64), ReLU ---
    zero_f32_kernel<<<ZB, ZT, 0, stream>>>(agg, (long long)N * D_IN);
    {
        long long total = (long long)E * (D_IN / 4);   // lgc = 5
        int blocks = (int)((total + 255) / 256);
        scatter_add_kernel<<<blocks, 256, 0, stream>>>(x, src, dst, agg, total, D_IN, 5);
    }
    {
        int ntn = D_HID / 16;                          // 4
        int tiles = (N / 16) * ntn;                    // 12500
        sage_wmma_kernel<<<(tiles + 3) / 4, 128, 0, stream>>>(agg, inv_deg, x, P1l, P1r,
                                                              bp1, h1, N, D_IN, D_HID, ntn, 1);
    }

    // --- layer 2: h1 (K=64) -> h2 (64), ReLU ---
    zero_f32_kernel<<<ZB, ZT, 0, stream>>>(agg, (long long)N * D_HID);
    {
        long long total = (long long)E * (D_HID / 4);  // lgc = 4
        int blocks = (int)((total + 255) / 256);
        scatter_add_kernel<<<blocks, 256, 0, stream>>>(h1, src, dst, agg, total, D_HID, 4);
    }
    {
        int ntn = D_HID / 16;
        int tiles = (N / 16) * ntn;
        sage_wmma_kernel<<<(tiles + 3) / 4, 128, 0, stream>>>(agg, inv_deg, h1, P2l, P2r,
                                                              bp2, h2, N, D_HID, D_HID, ntn, 1);
    }

    // --- layer 3: h2 (K=64) -> out (6), no ReLU ---
    zero_f32_kernel<<<ZB, ZT, 0, stream>>>(agg, (long long)N * D_HID);
    {
        long long total = (long long)E * (D_HID / 4);
        int blocks = (int)((total + 255) / 256);
        scatter_add_kernel<<<blocks, 256, 0, stream>>>(h2, src, dst, agg, total, D_HID, 4);
    }
    {
        int ntn = 1;                                   // Nout=6 padded into one 16-wide tile
        int tiles = (N / 16) * ntn;                    // 3125
        sage_wmma_kernel<<<(tiles + 3) / 4, 128, 0, stream>>>(agg, inv_deg, h2, P3l, P3r,
                                                              bp3, out, N, D_HID, D_OUT, ntn, 0);
    }
}